// ChunkStrategy_10720238370920
// MI455X (gfx1250) — compile-verified
//
#include <hip/hip_runtime.h>
#include <hip/hip_bf16.h>
#include <math.h>

// Problem constants (from reference setup_inputs)
#define BB      16
#define SS      768
#define HID     128
#define DE      16
#define ROW_F4  ((SS * DE) / 4)   // 3072 float4 per (b,s) row
#define NTILES  (SS / 16)         // 48 row-tiles per batch
#define NWAVES  8                 // 256 threads = 8 wave32

typedef __attribute__((ext_vector_type(2)))  float    v2f;
typedef __attribute__((ext_vector_type(4)))  float    f4;
typedef __attribute__((ext_vector_type(8)))  float    v8f;

// ---------------------------------------------------------------------------
// Kernel A: stream edge_features once; per (b,s) row produce
//   es[b,s]   = mean over (k,d) of edge[b,s,k,d]
//   dist[b,s] = min over pocket k of edge[b,s,k,0]   (+inf if no pocket)
// 604 MB streamed -> bandwidth bound; nontemporal loads (data > L2).
// ---------------------------------------------------------------------------
__global__ void __launch_bounds__(256) edge_reduce_kernel(
    const float* __restrict__ edge, const int* __restrict__ pocket,
    float* __restrict__ es_out, float* __restrict__ dist_out) {
  const int row = blockIdx.x;            // b*SS + s
  const int b   = row / SS;
  const int tid = threadIdx.x;

  __shared__ unsigned char pm[SS];
  __shared__ float rsum[256];
  __shared__ float rmin[256];

  for (int k = tid; k < SS; k += 256)
    pm[k] = (unsigned char)(pocket[b * SS + k] != 0);
  __syncthreads();

  const f4* __restrict__ p4 = (const f4*)(edge + (size_t)row * (SS * DE));
  float s  = 0.0f;
  float mn = __builtin_inff();
  #pragma unroll 4
  for (int i = tid; i < ROW_F4; i += 256) {
    f4 v = __builtin_nontemporal_load(p4 + i);
    s += v.x + v.y + v.z + v.w;
    if ((i & 3) == 0) {                  // v.x == edge[b,s,k=i>>2,0]
      if (pm[i >> 2]) mn = fminf(mn, v.x);
    }
  }
  rsum[tid] = s;
  rmin[tid] = mn;
  __syncthreads();
  for (int off = 128; off > 0; off >>= 1) {
    if (tid < off) {
      rsum[tid] += rsum[tid + off];
      rmin[tid]  = fminf(rmin[tid], rmin[tid + off]);
    }
    __syncthreads();
  }
  if (tid == 0) {
    es_out[row]   = rsum[0] * (1.0f / (float)(SS * DE));
    dist_out[row] = rmin[0];
  }
}

// ---------------------------------------------------------------------------
// Kernel B: one block (256 thr = 8 waves) per batch.
//  Phase 0: stage W_node in LDS; fold edge-MLP + biases into u[64], cconst.
//  Phase 1: per wave, 16-row tile: node_enc via v_wmma_f32_16x16x4_f32
//           (2 accumulator tiles, K=128 in 32 steps), dot with W_imp[0:32],
//           add closed-form edge term, sigmoid*mask -> imp[s] in LDS.
//  Phase 2: imp mean; Phase 3: merged count + chunk-size decision.
// ---------------------------------------------------------------------------
__global__ void __launch_bounds__(256) chunk_kernel(
    const float* __restrict__ nf,   const float* __restrict__ seqm,
    const int*   __restrict__ pocket,
    const float* __restrict__ Wn,   const float* __restrict__ bn,
    const float* __restrict__ We1,  const float* __restrict__ be1,
    const float* __restrict__ We2,  const float* __restrict__ be2,
    const float* __restrict__ Wimp, const float* __restrict__ bimp,
    const float* __restrict__ es_in, const float* __restrict__ dist_in,
    float* __restrict__ out, int out_size) {
  const int b    = blockIdx.x;
  const int tid  = threadIdx.x;
  const int lane = tid & 31;
  const int wv   = tid >> 5;
  const int m    = lane & 15;      // row (A) / col (B) index within tile
  const int half = lane >> 4;      // K sub-phase for 16x16x4 layout

  __shared__ float Wn_s[HID * 32];          // 16 KB: W_node staged
  __shared__ float we1_s[64], be1_s[64], u_s[64], wimp_s[32];
  __shared__ float imp_s[SS];
  __shared__ float part_s[NWAVES * 256];    // per-wave D.Wimp partials
  __shared__ float red_s[256];
  __shared__ int   redi_s[256];
  __shared__ int   redp_s[256];
  __shared__ float cconst_s;
  __shared__ float mean_s;

  // ---- Phase 0: weights into LDS, algebraic folds ----
  for (int i = tid; i < HID * 32; i += 256) Wn_s[i] = Wn[i];
  if (tid < 64) {
    float u = 0.0f;
    for (int j = 0; j < 32; ++j) u += We2[tid * 32 + j] * Wimp[32 + j];
    u_s[tid]   = u;
    we1_s[tid] = We1[tid];
    be1_s[tid] = be1[tid];
  }
  if (tid < 32) wimp_s[tid] = Wimp[tid];
  if (tid == 0) {
    float c = bimp[0];
    for (int j = 0; j < 32; ++j) c += be2[j] * Wimp[32 + j] + bn[j] * Wimp[j];
    cconst_s = c;
  }
  __syncthreads();

  // ---- Phase 1: WMMA tiles ----
  for (int t = wv; t < NTILES; t += NWAVES) {
    const int s0 = t * 16;
    const float* __restrict__ Arow = nf + ((size_t)b * SS + s0 + m) * HID;
    v8f acc0 = {0.f,0.f,0.f,0.f,0.f,0.f,0.f,0.f};
    v8f acc1 = {0.f,0.f,0.f,0.f,0.f,0.f,0.f,0.f};

    for (int k0 = 0; k0 < HID; k0 += 4) {
      const int kk = k0 + 2 * half;  // lanes 0-15: K=k0,k0+1; 16-31: k0+2,k0+3
      const float2 av = *(const float2*)(Arow + kk);
      v2f a;  a.x  = av.x;                     a.y  = av.y;
      v2f b0; b0.x = Wn_s[kk * 32 + m];        b0.y = Wn_s[(kk + 1) * 32 + m];
      v2f b1; b1.x = Wn_s[kk * 32 + 16 + m];   b1.y = Wn_s[(kk + 1) * 32 + 16 + m];
      acc0 = __builtin_amdgcn_wmma_f32_16x16x4_f32(false, a, false, b0,
                                                   (short)0, acc0, false, false);
      acc1 = __builtin_amdgcn_wmma_f32_16x16x4_f32(false, a, false, b1,
                                                   (short)0, acc1, false, false);
    }

    // D layout: VGPR j = row (j or j+8 by lane half), col = 16*tile + m.
    // partial[j] = D[row_j][m]*Wimp[m] + D[row_j][16+m]*Wimp[16+m]
    const float wA = wimp_s[m];
    const float wB = wimp_s[16 + m];
    float* pw = part_s + wv * 256;
    #pragma unroll
    for (int j = 0; j < 8; ++j)
      pw[j * 32 + lane] = acc0[j] * wA + acc1[j] * wB;
    __builtin_amdgcn_wave_barrier();   // same-wave LDS ordering

    if (lane < 16) {
      const int r = lane;              // tile row 0..15
      const float* pr = pw + (r & 7) * 32 + (r >> 3) * 16;
      float rowsum = 0.0f;
      #pragma unroll
      for (int l = 0; l < 16; ++l) rowsum += pr[l];

      const int   s   = s0 + r;
      const float esv = es_in[b * SS + s];
      float lg = rowsum + cconst_s;    // node dot + all folded biases
      for (int i = 0; i < 64; ++i)     // closed-form edge-MLP contribution
        lg += fmaxf(fmaf(esv, we1_s[i], be1_s[i]), 0.0f) * u_s[i];
      imp_s[s] = seqm[b * SS + s] / (1.0f + __expf(-lg));
    }
    __builtin_amdgcn_wave_barrier();
  }
  __syncthreads();

  // ---- Phase 2: imp mean (== imp_full, same reduction) ----
  float ps = 0.0f;
  for (int s = tid; s < SS; s += 256) ps += imp_s[s];
  red_s[tid] = ps;
  __syncthreads();
  for (int off = 128; off > 0; off >>= 1) {
    if (tid < off) red_s[tid] += red_s[tid + off];
    __syncthreads();
  }
  if (tid == 0) mean_s = red_s[0] / (float)SS;
  __syncthreads();
  const float mean = mean_s;

  // ---- Phase 3: merged count + decision ----
  int cnt = 0, anyp = 0;
  for (int s = tid; s < SS; s += 256) {
    const int   p = (pocket[b * SS + s] != 0);
    const float d = dist_in[b * SS + s];
    const int core  = d < 6.0f;
    const int shell = (d >= 6.0f) & (d < 10.0f) & (imp_s[s] > mean);
    cnt  += (p | core | shell);
    anyp |= p;
  }
  redi_s[tid] = cnt;
  redp_s[tid] = anyp;
  __syncthreads();
  for (int off = 128; off > 0; off >>= 1) {
    if (tid < off) {
      redi_s[tid] += redi_s[tid + off];
      redp_s[tid] |= redp_s[tid + off];
    }
    __syncthreads();
  }
  if (tid == 0) {
    float chunk;
    if (redp_s[0]) {                       // pocket branch (adj = 64)
      int ml = redi_s[0];
      ml = ml < 256 ? ml : 256;            // merged_len clamp
      ml = ml > 64  ? ml : 64;             // max(., adj)
      chunk = (float)(ml < 256 ? ml : 256);
    } else {                               // no-pocket branch
      chunk = fminf(fmaxf(64.0f * mean, 32.0f), 128.0f);
    }
    out[b] = chunk;
    if (b == 0 && out_size > BB) out[BB] = 256.0f;  // MAX_SEQ_LEN
  }
}

// ---------------------------------------------------------------------------
extern "C" void kernel_launch(void* const* d_in, const int* in_sizes, int n_in,
                              void* d_out, int out_size, void* d_ws, size_t ws_size,
                              hipStream_t stream) {
  const float* nf   = (const float*)d_in[0];   // [16,768,128]
  const float* ef   = (const float*)d_in[1];   // [16,768,768,16]
  const float* sm   = (const float*)d_in[2];   // [16,768]
  const int*   pk   = (const int*)  d_in[3];   // [16,768] bool->int
  const float* Wn   = (const float*)d_in[4];   // [128,32]
  const float* bn   = (const float*)d_in[5];   // [32]
  const float* We1  = (const float*)d_in[6];   // [1,64]
  const float* be1  = (const float*)d_in[7];   // [64]
  const float* We2  = (const float*)d_in[8];   // [64,32]
  const float* be2  = (const float*)d_in[9];   // [32]
  const float* Wimp = (const float*)d_in[10];  // [64,1]
  const float* bimp = (const float*)d_in[11];  // [1]
  float* out = (float*)d_out;

  float* es   = (float*)d_ws;        // [16*768]
  float* dist = es + BB * SS;        // [16*768]

  edge_reduce_kernel<<<BB * SS, 256, 0, stream>>>(ef, pk, es, dist);
  chunk_kernel<<<BB, 256, 0, stream>>>(nf, sm, pk, Wn, bn, We1, be1,
                                       We2, be2, Wimp, bimp,
                                       es, dist, out, out_size);
}